// GCNGraphClassifierIMDB_20461224198761
// MI455X (gfx1250) — compile-verified
//
#include <hip/hip_runtime.h>
#include <hip/hip_bf16.h>
#include <math.h>

typedef __attribute__((ext_vector_type(2))) float v2f;
typedef __attribute__((ext_vector_type(8))) float v8f;

#define N_NODES   100000
#define N_FEAT    128
#define HIDDEN    16
#define N_CLASSES 2
#define N_GRAPHS  5000

// ---------------------------------------------------------------------------
// 1) init: deg (stored in `dis` slot) = 1.0 (self loop), agg = 0, pooled = 0,
//    cnt = 0.  Runs every call so graph replay is deterministic.
// ---------------------------------------------------------------------------
__global__ void k_init(float* __restrict__ dis, float* __restrict__ agg,
                       float* __restrict__ pooled, float* __restrict__ cnt,
                       int n, int g) {
    int i = blockIdx.x * blockDim.x + threadIdx.x;
    if (i < n * HIDDEN) agg[i] = 0.0f;
    if (i < n)          dis[i] = 1.0f;          // self-loop contributes 1 to deg
    if (i < g * HIDDEN) pooled[i] = 0.0f;
    if (i < g)          cnt[i] = 0.0f;
}

// ---------------------------------------------------------------------------
// 2) degree accumulation over edges (deg lives in `dis` buffer)
// ---------------------------------------------------------------------------
__global__ void k_degree(const int* __restrict__ edge_dst, float* __restrict__ deg,
                         int e) {
    int i = blockIdx.x * blockDim.x + threadIdx.x;
    if (i >= e) return;
    atomicAdd(&deg[edge_dst[i]], 1.0f);
}

// ---------------------------------------------------------------------------
// 3) dis = rsqrt(deg)   (in place)
// ---------------------------------------------------------------------------
__global__ void k_rsqrt(float* __restrict__ dis, int n) {
    int i = blockIdx.x * blockDim.x + threadIdx.x;
    if (i >= n) return;
    dis[i] = rsqrtf(dis[i]);
}

// ---------------------------------------------------------------------------
// 4/7) H = X @ W using V_WMMA_F32_16X16X4_F32.
//      One wave per 16-row output tile; W is K x 16 row-major.
//      A frag: lane L -> M = L%16, K = 2*(L>=16)+j   (j = vector elem)
//      B frag: lane L -> N = L%16, K = 2*(L>=16)+j
//      D     : lane L, acc v -> M = v + 8*(L>=16), N = L%16
// ---------------------------------------------------------------------------
template <int K>
__global__ void k_gemm_wmma(const float* __restrict__ X, const float* __restrict__ W,
                            float* __restrict__ H, int nTiles) {
    int wave = (blockIdx.x * blockDim.x + threadIdx.x) >> 5;
    if (wave >= nTiles) return;                 // wave-uniform: EXEC stays all-1s
    int lane = threadIdx.x & 31;
    int m  = lane & 15;                         // M (for A) / N (for B, D)
    int kh = (lane >> 4) << 1;                  // K sub-offset: 0 or 2

    const float* xrow = X + (size_t)(wave * 16 + m) * K;
    v8f c = {};
#pragma unroll
    for (int k = 0; k < K; k += 4) {
        v2f a, b;
        a[0] = xrow[k + kh];
        a[1] = xrow[k + kh + 1];
        b[0] = W[(k + kh) * HIDDEN + m];
        b[1] = W[(k + kh + 1) * HIDDEN + m];
        c = __builtin_amdgcn_wmma_f32_16x16x4_f32(
                /*neg_a=*/false, a, /*neg_b=*/false, b,
                /*c_mod=*/(short)0, c, /*reuse_a=*/false, /*reuse_b=*/false);
    }
    float* hrow = H + (size_t)wave * 16 * HIDDEN;
    int mbase = (lane >> 4) * 8;
#pragma unroll
    for (int v = 0; v < 8; ++v)
        hrow[(mbase + v) * HIDDEN + m] = c[v];
}

// ---------------------------------------------------------------------------
// 5/8) edge scatter: agg[dst] += h[src] * dis[src] * dis[dst]
//      16 lanes handle the 16 features of one edge -> contiguous 64B gather
//      and contiguous atomic adds.
// ---------------------------------------------------------------------------
__global__ void k_scatter(const int* __restrict__ edge_src, const int* __restrict__ edge_dst,
                          const float* __restrict__ dis, const float* __restrict__ h,
                          float* __restrict__ agg, int e) {
    int gid = blockIdx.x * blockDim.x + threadIdx.x;
    int eid = gid >> 4;
    if (eid >= e) return;
    int f   = gid & 15;
    int s   = edge_src[eid];
    int d   = edge_dst[eid];
    float norm = dis[s] * dis[d];
    float val  = h[(size_t)s * HIDDEN + f] * norm;
    atomicAdd(&agg[(size_t)d * HIDDEN + f], val);
}

// ---------------------------------------------------------------------------
// 6/9) fuse self-loop term + bias + ReLU; re-zero agg for next layer
//      out = relu(agg + h * dis^2 + bias)
// ---------------------------------------------------------------------------
__global__ void k_self_relu(const float* __restrict__ h, float* __restrict__ agg,
                            float* __restrict__ out, const float* __restrict__ dis,
                            const float* __restrict__ bias, int n) {
    int i = blockIdx.x * blockDim.x + threadIdx.x;
    if (i >= n * HIDDEN) return;
    int node = i >> 4, f = i & 15;
    float d = dis[node];
    float v = agg[i] + h[i] * d * d + bias[f];
    agg[i] = 0.0f;                              // ready for next scatter pass
    out[i] = v > 0.0f ? v : 0.0f;
}

// ---------------------------------------------------------------------------
// 10) mean-pool accumulation by graph id
// ---------------------------------------------------------------------------
__global__ void k_pool(const int* __restrict__ batch, const float* __restrict__ out,
                       float* __restrict__ pooled, float* __restrict__ cnt, int n) {
    int i = blockIdx.x * blockDim.x + threadIdx.x;
    if (i >= n * HIDDEN) return;
    int node = i >> 4, f = i & 15;
    int b = batch[node];
    atomicAdd(&pooled[(size_t)b * HIDDEN + f], out[i]);
    if (f == 0) atomicAdd(&cnt[b], 1.0f);
}

// ---------------------------------------------------------------------------
// 11) head: logits = (pooled/cnt) @ Wfc + bfc ; log_softmax over 2 classes
// ---------------------------------------------------------------------------
__global__ void k_head(const float* __restrict__ pooled, const float* __restrict__ cnt,
                       const float* __restrict__ Wfc, const float* __restrict__ bfc,
                       float* __restrict__ y, int g) {
    int i = blockIdx.x * blockDim.x + threadIdx.x;
    if (i >= g) return;
    float c = fmaxf(cnt[i], 1.0f);
    float l0 = bfc[0], l1 = bfc[1];
#pragma unroll
    for (int j = 0; j < HIDDEN; ++j) {
        float p = pooled[(size_t)i * HIDDEN + j] / c;
        l0 += p * Wfc[j * N_CLASSES + 0];
        l1 += p * Wfc[j * N_CLASSES + 1];
    }
    float m  = fmaxf(l0, l1);
    float ls = m + logf(expf(l0 - m) + expf(l1 - m));
    y[i * 2 + 0] = l0 - ls;
    y[i * 2 + 1] = l1 - ls;
}

// ---------------------------------------------------------------------------
extern "C" void kernel_launch(void* const* d_in, const int* in_sizes, int n_in,
                              void* d_out, int out_size, void* d_ws, size_t ws_size,
                              hipStream_t stream) {
    const float* x     = (const float*)d_in[0];   // [N, 128]
    const int*   eidx  = (const int*)  d_in[1];   // [2, E] flat
    const int*   batch = (const int*)  d_in[2];   // [N]
    const float* W1    = (const float*)d_in[3];   // [128, 16]
    const float* b1    = (const float*)d_in[4];
    const float* W2    = (const float*)d_in[5];   // [16, 16]
    const float* b2    = (const float*)d_in[6];
    const float* Wfc   = (const float*)d_in[7];   // [16, 2]
    const float* bfc   = (const float*)d_in[8];
    float* y = (float*)d_out;

    const int n = N_NODES;
    const int e = in_sizes[1] / 2;
    const int g = N_GRAPHS;
    const int* esrc = eidx;
    const int* edst = eidx + e;

    // workspace layout (floats): dis[n] | h[n*16] | agg[n*16] | out[n*16]
    //                            | pooled[g*16] | cnt[g]   (~20 MB total)
    float* ws     = (float*)d_ws;
    float* dis    = ws;
    float* h      = dis + n;
    float* agg    = h + (size_t)n * HIDDEN;
    float* outb   = agg + (size_t)n * HIDDEN;
    float* pooled = outb + (size_t)n * HIDDEN;
    float* cnt    = pooled + (size_t)g * HIDDEN;

    const int BLK = 256;
    int nh_blocks   = (n * HIDDEN + BLK - 1) / BLK;
    int e_blocks    = (e + BLK - 1) / BLK;
    int n_blocks    = (n + BLK - 1) / BLK;
    int ef_blocks   = ((e * 16) + BLK - 1) / BLK;   // e*16 = 51.2M threads
    int nTiles      = n / 16;                        // 100000 % 16 == 0
    int gemm_blocks = (nTiles * 32 + BLK - 1) / BLK;
    int g_blocks    = (g + BLK - 1) / BLK;

    // 1) init
    k_init<<<nh_blocks, BLK, 0, stream>>>(dis, agg, pooled, cnt, n, g);
    // 2) degree (into dis buffer), 3) rsqrt
    k_degree<<<e_blocks, BLK, 0, stream>>>(edst, dis, e);
    k_rsqrt<<<n_blocks, BLK, 0, stream>>>(dis, n);

    // layer 1: h = x @ W1 ; scatter ; relu(agg + self + b1) -> outb
    k_gemm_wmma<N_FEAT><<<gemm_blocks, BLK, 0, stream>>>(x, W1, h, nTiles);
    k_scatter<<<ef_blocks, BLK, 0, stream>>>(esrc, edst, dis, h, agg, e);
    k_self_relu<<<nh_blocks, BLK, 0, stream>>>(h, agg, outb, dis, b1, n);

    // layer 2: h = outb @ W2 ; scatter ; relu -> outb (in place, new pass)
    k_gemm_wmma<HIDDEN><<<gemm_blocks, BLK, 0, stream>>>(outb, W2, h, nTiles);
    k_scatter<<<ef_blocks, BLK, 0, stream>>>(esrc, edst, dis, h, agg, e);
    k_self_relu<<<nh_blocks, BLK, 0, stream>>>(h, agg, outb, dis, b2, n);

    // pooling + classifier head
    k_pool<<<nh_blocks, BLK, 0, stream>>>(batch, outb, pooled, cnt, n);
    k_head<<<g_blocks, BLK, 0, stream>>>(pooled, cnt, Wfc, bfc, y, g);
    (void)out_size; (void)ws_size; (void)n_in;
}